// loss_54142357734099
// MI455X (gfx1250) — compile-verified
//
#include <hip/hip_runtime.h>
#include <float.h>

typedef __attribute__((ext_vector_type(2))) float v2f;
typedef __attribute__((ext_vector_type(8))) float v8f;

#define NUM_PTS 1024
#define LATENT_SIZE 32
#define BETA 0.1f

__global__ void zero_out_kernel(float* out) { out[0] = 0.0f; }

// One wave (32 lanes) per (batch, direction, 16-row tile of the "A" cloud).
// dir 0: A = x, B = recon  -> per-x-point min over recon points (loss_2)
// dir 1: A = recon, B = x  -> per-recon-point min over x points (loss_1)
// P[i,j] = xx[i] + yy[j] - 2 * <A_i, B_j>; xx[i] added after the min.
__global__ __launch_bounds__(256) void chamfer_kernel(const float* __restrict__ recon,
                                                      const float* __restrict__ x,
                                                      float* __restrict__ out)
{
    const int lane    = threadIdx.x & 31;
    const int wave    = threadIdx.x >> 5;            // 0..7
    const int rowTile = blockIdx.x * 8 + wave;       // 0..63
    const int b       = blockIdx.y;                  // batch
    const int dir     = blockIdx.z;                  // 0 or 1

    const float* Ap = dir ? recon : x;
    const float* Bp = dir ? x : recon;

    const int half = lane >> 4;   // 0: holds K=0,1  1: holds K=2,(pad 0)
    const int l16  = lane & 15;

    // ---- Load A tile (16 rows x 4(K), K=3 padded to 4) ----
    // 32-bit A-matrix 16x4 layout: lanes 0-15 = rows, VGPR0/1 = K0/K1;
    // lanes 16-31 = rows, VGPR0/1 = K2/K3.
    const float* ap = Ap + ((size_t)b * NUM_PTS + (size_t)rowTile * 16 + l16) * 3;
    v2f av;
    av.x = half ? ap[2] : ap[0];
    av.y = half ? 0.0f  : ap[1];

    // Per-lane partial squared norm; wave-sum later gives sum_rows xx.
    float sq = av.x * av.x + av.y * av.y;

    float rmin[8];
#pragma unroll
    for (int v = 0; v < 8; ++v) rmin[v] = FLT_MAX;

    for (int ct = 0; ct < 64; ++ct) {
        // ---- Load B tile (4(K) x 16 cols): same lane pattern as A ----
        const float* bp = Bp + ((size_t)b * NUM_PTS + (size_t)ct * 16 + l16) * 3;
        float r0 = bp[0], r1 = bp[1], r2 = bp[2];
        v2f bv;
        bv.x = half ? r2   : r0;
        bv.y = half ? 0.0f : r1;
        float yy = r0 * r0 + r1 * r1 + r2 * r2;  // ||B_col||^2 for col = ct*16 + l16

        // D = A x B : 16x16 f32 tile of inner products, in 8 VGPRs.
        v8f c = {};
        c = __builtin_amdgcn_wmma_f32_16x16x4_f32(
                /*neg_a=*/false, av, /*neg_b=*/false, bv,
                /*c_mod=*/(short)0, c, /*reuse_a=*/false, /*reuse_b=*/false);

        // Lane holds column (l16); VGPR v holds row v + 8*half.
#pragma unroll
        for (int v = 0; v < 8; ++v) {
            float val = yy - 2.0f * c[v];
            rmin[v] = fminf(rmin[v], val);
        }
    }

    // Min-reduce each row across the 16 lanes of this half (columns).
#pragma unroll
    for (int v = 0; v < 8; ++v) {
#pragma unroll
        for (int m = 1; m <= 8; m <<= 1)
            rmin[v] = fminf(rmin[v], __shfl_xor(rmin[v], m, 32));
    }
    float s = 0.0f;
#pragma unroll
    for (int v = 0; v < 8; ++v) s += rmin[v];   // rows {0..7} or {8..15}
    s += __shfl_xor(s, 16, 32);                 // combine halves -> all 16 rows

    // Wave-sum sq over all 32 lanes == sum over the 16 rows of xx.
#pragma unroll
    for (int m = 1; m <= 16; m <<= 1) sq += __shfl_xor(sq, m, 32);

    if (lane == 0) {
        atomicAdd(out, (s + sq) * (1.0f / (float)NUM_PTS));
    }
}

__global__ __launch_bounds__(256) void kld_kernel(const float* __restrict__ mu,
                                                  const float* __restrict__ logvar,
                                                  float* __restrict__ out, int n)
{
    int i = blockIdx.x * blockDim.x + threadIdx.x;
    float t = 0.0f;
    if (i < n) {
        float lv = logvar[i], m = mu[i];
        t = 1.0f + lv - m * m - __expf(lv);
    }
#pragma unroll
    for (int m = 1; m <= 16; m <<= 1) t += __shfl_xor(t, m, 32);
    if ((threadIdx.x & 31) == 0)
        atomicAdd(out, t * (-0.5f * BETA / (float)LATENT_SIZE));
}

extern "C" void kernel_launch(void* const* d_in, const int* in_sizes, int n_in,
                              void* d_out, int out_size, void* d_ws, size_t ws_size,
                              hipStream_t stream) {
    const float* recon  = (const float*)d_in[0];
    const float* x      = (const float*)d_in[1];
    const float* mu     = (const float*)d_in[2];
    const float* logvar = (const float*)d_in[3];
    float* out = (float*)d_out;

    const int B = in_sizes[0] / (NUM_PTS * 3);        // 128
    const int nlat = in_sizes[2];                     // B * LATENT_SIZE

    zero_out_kernel<<<1, 1, 0, stream>>>(out);

    dim3 grid(64 / 8, B, 2);                          // 8 waves/block, 64 row tiles
    chamfer_kernel<<<grid, 256, 0, stream>>>(recon, x, out);

    kld_kernel<<<(nlat + 255) / 256, 256, 0, stream>>>(mu, logvar, out, nlat);
}